// GraphNeuralNetwork_1683627180352
// MI455X (gfx1250) — compile-verified
//
#include <hip/hip_runtime.h>

typedef __attribute__((ext_vector_type(2))) float v2f;
typedef __attribute__((ext_vector_type(8))) float v8f;

// Async global->LDS staging if the toolchain declares the gfx1250 builtins.
#if defined(__has_builtin)
#  if __has_builtin(__builtin_amdgcn_global_load_async_to_lds_b32) && \
      __has_builtin(__builtin_amdgcn_s_wait_asynccnt)
#    define ASYNC_LDS 1
#  endif
#endif
#ifndef ASYNC_LDS
#  define ASYNC_LDS 0
#endif

typedef __attribute__((address_space(1))) int* gptr_i32;
typedef __attribute__((address_space(3))) int* lptr_i32;

// ---------------------------------------------------------------------------
// Utility kernels
// ---------------------------------------------------------------------------
__global__ void zero_f32(float* __restrict__ p, long n) {
    long i = (long)blockIdx.x * blockDim.x + threadIdx.x;
    long stride = (long)gridDim.x * blockDim.x;
    for (; i < n; i += stride) p[i] = 0.0f;
}

__global__ void degree_kernel(const int* __restrict__ dst, float* __restrict__ deg, int E) {
    int i = blockIdx.x * blockDim.x + threadIdx.x;
    if (i < E) atomicAdd(deg + dst[i], 1.0f);
}

__global__ void invdeg_kernel(float* __restrict__ deg, int n) {
    int i = blockIdx.x * blockDim.x + threadIdx.x;
    if (i < n) deg[i] = 1.0f / fmaxf(deg[i], 1.0f);
}

// One wave (32 lanes) per edge: lane l moves channels 2l, 2l+1.
// float2 gather (coalesced 256B per wave) + 2x global_atomic_add_f32.
__global__ void __launch_bounds__(256)
scatter_add_kernel(const float* __restrict__ h, const int* __restrict__ dst,
                   const int* __restrict__ src, float* agg, int E) {
    int e = blockIdx.x * (blockDim.x >> 5) + (threadIdx.x >> 5);
    int lane = threadIdx.x & 31;
    if (e >= E) return;                        // wave-uniform exit
    int d = dst[e];
    int s = src[e];
    float2 v = *((const float2*)(h + (long)s * 64) + lane);
    float* ap = agg + (long)d * 64 + lane * 2;
    atomicAdd(ap, v.x);
    atomicAdd(ap + 1, v.y);
}

// ---------------------------------------------------------------------------
// Weight staging: global [K,64] row-major -> LDS transposed wt[n*(K+2)+k].
// Padding of 2 words gives conflict-free ds_load_b64 B-fragment reads.
// Uses per-lane GLOBAL_LOAD_ASYNC_TO_LDS_B32 (ASYNCcnt) when available.
// ---------------------------------------------------------------------------
__device__ __forceinline__ void stage_weight(const float* __restrict__ W,
                                             float* wt, int K, int ldk) {
    for (int i = threadIdx.x; i < K * 64; i += 256) {
        int k = i >> 6;          // row in W  (K dim)
        int n = i & 63;          // col in W  (N dim)
#if ASYNC_LDS
        __builtin_amdgcn_global_load_async_to_lds_b32(
            (gptr_i32)(W + i),
            (lptr_i32)(wt + n * ldk + k), 0, 0);
#else
        wt[n * ldk + k] = W[i];
#endif
    }
}

__device__ __forceinline__ void stage_fence() {
#if ASYNC_LDS
    __builtin_amdgcn_s_wait_asynccnt(0);
#endif
    __syncthreads();
}

// ---------------------------------------------------------------------------
// WMMA fp32 GEMM: one wave computes a 16-row x 64-col tile of Out = A[K] @ W[K,64]
// A fragment (16x4 f32): lanes 0-15 -> K pair {kk,kk+1}, lanes 16-31 -> {kk+2,kk+3}
// B fragment (4x16 f32): same split along K, col = lane%16  (ds_load_b64 from LDS)
// C/D (16x16 f32): vgpr r, lanes 0-15 -> row r, lanes 16-31 -> row r+8
// ---------------------------------------------------------------------------
#define WMMA_F32(a, b, c) \
    __builtin_amdgcn_wmma_f32_16x16x4_f32(false, (a), false, (b), (short)0, (c), false, false)

template <int K>
__global__ void __launch_bounds__(256)
gemm_wmma_kernel(const float* __restrict__ A, const float* __restrict__ W,
                 float* __restrict__ Out, int nRows) {
    constexpr int LDK = K + 2;
    __shared__ float wt[64 * LDK];
    stage_weight(W, wt, K, LDK);
    stage_fence();                             // all waves participate, then exit

    int nTiles = (nRows + 15) >> 4;
    int tile = blockIdx.x * (blockDim.x >> 5) + (threadIdx.x >> 5);
    if (tile >= nTiles) return;                // wave-uniform: EXEC stays all-1s
    int lane = threadIdx.x & 31;
    int half = lane >> 4;                      // 0 | 1
    int sub  = lane & 15;

    long rowA = (long)tile * 16 + sub;
    if (rowA >= nRows) rowA = nRows - 1;       // safe clamp (N%16==0 in practice)
    const float* arow = A + rowA * K;
    const float* wl = wt + sub * LDK + 2 * half;   // this lane's K-pair column base

    v8f acc0 = {}, acc1 = {}, acc2 = {}, acc3 = {};
#pragma unroll
    for (int kk = 0; kk < K; kk += 4) {
        v2f a;
        a.x = arow[kk + 2 * half];
        a.y = arow[kk + 2 * half + 1];
        v2f b0 = *(const v2f*)(wl + kk);
        v2f b1 = *(const v2f*)(wl + 16 * LDK + kk);
        v2f b2 = *(const v2f*)(wl + 32 * LDK + kk);
        v2f b3 = *(const v2f*)(wl + 48 * LDK + kk);
        acc0 = WMMA_F32(a, b0, acc0);
        acc1 = WMMA_F32(a, b1, acc1);
        acc2 = WMMA_F32(a, b2, acc2);
        acc3 = WMMA_F32(a, b3, acc3);
    }

    long rowO = (long)tile * 16 + half * 8;
#pragma unroll
    for (int r = 0; r < 8; r++) {
        long row = rowO + r;
        if (row < nRows) {
            float* o = Out + row * 64 + sub;
            o[ 0] = acc0[r];
            o[16] = acc1[r];
            o[32] = acc2[r];
            o[48] = acc3[r];
        }
    }
}

// Fused combine: out = relu( (inv_deg[row] * agg) @ Wagg + h @ Wself + bias )
// In-place out==agg is safe: a wave only touches its own 16 rows, stores
// data-depend on all prior loads. (agg/out deliberately NOT restrict.)
__global__ void __launch_bounds__(256)
combine_wmma_kernel(const float* agg, const float* __restrict__ h,
                    const float* __restrict__ Wagg, const float* __restrict__ Wself,
                    const float* __restrict__ bias, const float* __restrict__ invdeg,
                    float* out, int nRows) {
    constexpr int K = 64, LDK = K + 2;
    __shared__ float wta[64 * LDK];
    __shared__ float wts[64 * LDK];
    stage_weight(Wagg,  wta, K, LDK);
    stage_weight(Wself, wts, K, LDK);
    stage_fence();

    int nTiles = (nRows + 15) >> 4;
    int tile = blockIdx.x * (blockDim.x >> 5) + (threadIdx.x >> 5);
    if (tile >= nTiles) return;
    int lane = threadIdx.x & 31;
    int half = lane >> 4;
    int sub  = lane & 15;

    long rowA = (long)tile * 16 + sub;
    if (rowA >= nRows) rowA = nRows - 1;
    const float rs = invdeg[rowA];             // fold mean-aggregation into GEMM
    const float* arow = agg + rowA * 64;
    const float* hrow = h   + rowA * 64;
    const float* wla = wta + sub * LDK + 2 * half;
    const float* wls = wts + sub * LDK + 2 * half;

    v8f acc0 = {}, acc1 = {}, acc2 = {}, acc3 = {};
#pragma unroll
    for (int kk = 0; kk < K; kk += 4) {
        // agg @ Wagg (row-scaled by 1/deg)
        v2f a;
        a.x = arow[kk + 2 * half] * rs;
        a.y = arow[kk + 2 * half + 1] * rs;
        v2f b0 = *(const v2f*)(wla + kk);
        v2f b1 = *(const v2f*)(wla + 16 * LDK + kk);
        v2f b2 = *(const v2f*)(wla + 32 * LDK + kk);
        v2f b3 = *(const v2f*)(wla + 48 * LDK + kk);
        acc0 = WMMA_F32(a, b0, acc0);
        acc1 = WMMA_F32(a, b1, acc1);
        acc2 = WMMA_F32(a, b2, acc2);
        acc3 = WMMA_F32(a, b3, acc3);
        // h @ Wself
        v2f c;
        c.x = hrow[kk + 2 * half];
        c.y = hrow[kk + 2 * half + 1];
        v2f s0 = *(const v2f*)(wls + kk);
        v2f s1 = *(const v2f*)(wls + 16 * LDK + kk);
        v2f s2 = *(const v2f*)(wls + 32 * LDK + kk);
        v2f s3 = *(const v2f*)(wls + 48 * LDK + kk);
        acc0 = WMMA_F32(c, s0, acc0);
        acc1 = WMMA_F32(c, s1, acc1);
        acc2 = WMMA_F32(c, s2, acc2);
        acc3 = WMMA_F32(c, s3, acc3);
    }

    float b0 = bias[sub], b1 = bias[sub + 16], b2 = bias[sub + 32], b3 = bias[sub + 48];
    long rowO = (long)tile * 16 + half * 8;
#pragma unroll
    for (int r = 0; r < 8; r++) {
        long row = rowO + r;
        if (row < nRows) {
            float* o = out + row * 64 + sub;
            o[ 0] = fmaxf(acc0[r] + b0, 0.0f);
            o[16] = fmaxf(acc1[r] + b1, 0.0f);
            o[32] = fmaxf(acc2[r] + b2, 0.0f);
            o[48] = fmaxf(acc3[r] + b3, 0.0f);
        }
    }
}

// ---------------------------------------------------------------------------
// Launch
// ---------------------------------------------------------------------------
extern "C" void kernel_launch(void* const* d_in, const int* in_sizes, int n_in,
                              void* d_out, int out_size, void* d_ws, size_t ws_size,
                              hipStream_t stream) {
    const float* x      = (const float*)d_in[0];   // [N,128]
    const int*   edge   = (const int*)  d_in[1];   // [2,E] row-major: dst then src
    const float* W_init = (const float*)d_in[2];   // [128,64]
    const float* W_self = (const float*)d_in[3];   // [2,64,64]
    const float* W_agg  = (const float*)d_in[4];   // [2,64,64]
    const float* bvec   = (const float*)d_in[5];   // [2,64]
    const float* W_out  = (const float*)d_in[6];   // [64,64]
    float* out = (float*)d_out;

    const int N = in_sizes[0] / 128;
    const int E = in_sizes[1] / 2;
    const int* dst = edge;
    const int* src = edge + E;

    float* deg = (float*)d_ws;                 // N floats -> becomes inv_deg
    float* hA  = deg + N;                      // N*64
    float* hB  = hA + (long)N * 64;            // N*64

    const int nTiles = (N + 15) / 16;
    const int gBlocks = (nTiles + 7) / 8;      // 8 waves (256 thr) per block

    // degree + reciprocal (mean aggregation)
    zero_f32<<<256, 256, 0, stream>>>(deg, N);
    degree_kernel<<<(E + 255) / 256, 256, 0, stream>>>(dst, deg, E);
    invdeg_kernel<<<(N + 255) / 256, 256, 0, stream>>>(deg, N);

    // h0 = x @ W_init
    gemm_wmma_kernel<128><<<gBlocks, 256, 0, stream>>>(x, W_init, hA, N);

    // layer 0: agg(hA) -> hB, combine -> hB
    zero_f32<<<2048, 256, 0, stream>>>(hB, (long)N * 64);
    scatter_add_kernel<<<(E + 7) / 8, 256, 0, stream>>>(hA, dst, src, hB, E);
    combine_wmma_kernel<<<gBlocks, 256, 0, stream>>>(hB, hA, W_agg, W_self,
                                                     bvec, deg, hB, N);

    // layer 1: agg(hB) -> hA, combine -> hA
    zero_f32<<<2048, 256, 0, stream>>>(hA, (long)N * 64);
    scatter_add_kernel<<<(E + 7) / 8, 256, 0, stream>>>(hB, dst, src, hA, E);
    combine_wmma_kernel<<<gBlocks, 256, 0, stream>>>(hA, hB, W_agg + 64 * 64,
                                                     W_self + 64 * 64, bvec + 64,
                                                     deg, hA, N);

    // out = h @ W_out
    gemm_wmma_kernel<64><<<gBlocks, 256, 0, stream>>>(hA, W_out, out, N);
}